// RGCN_16252156248487
// MI455X (gfx1250) — compile-verified
//
#include <hip/hip_runtime.h>

typedef float v2f __attribute__((ext_vector_type(2)));
typedef float v8f __attribute__((ext_vector_type(8)));

#define K_DIM 128   // both layers contract over 128

// ---------------- degree counting ----------------
__global__ void count_deg(const int* __restrict__ src, const int* __restrict__ dst,
                          int* __restrict__ cnt_out, int* __restrict__ cnt_in,
                          int E, int N) {
    int e = blockIdx.x * blockDim.x + threadIdx.x;
    int r = blockIdx.y;
    if (e >= E) return;
    int i = r * E + e;
    atomicAdd(&cnt_out[r * N + src[i]], 1);
    atomicAdd(&cnt_in [r * N + dst[i]], 1);
}

// counts (int) -> rsqrt(max(count,1)) stored as float bits, in place
__global__ void to_rsqrt(int* __restrict__ buf, int n) {
    int i = blockIdx.x * blockDim.x + threadIdx.x;
    if (i >= n) return;
    int c = buf[i];
    float f = rsqrtf((float)(c < 1 ? 1 : c));
    buf[i] = __float_as_int(f);
}

// out[n, D] = b[0,:] + b[1,:] + b[2,:]
__global__ void bias_init(float* __restrict__ out, const float* __restrict__ b,
                          long long total, int mask, int D) {
    long long i = (long long)blockIdx.x * blockDim.x + threadIdx.x;
    if (i >= total) return;
    int c = (int)i & mask;
    out[i] = b[c] + b[D + c] + b[2 * D + c];
}

// ---------------- WMMA GEMM: H = (act(X) * scale[:,None]) @ W ----------------
// X: [N,128], W: [128,Dout], H: [N,Dout]. One wave per 16x16 output tile.
__global__ __launch_bounds__(32)
void gemm_wmma_f32(const float* __restrict__ X, const float* __restrict__ W,
                   const float* __restrict__ scale, float* __restrict__ H,
                   int N, int Dout, int relu) {
    __shared__ float As[16 * 132];   // 16 rows x 128 (+4 pad) of scaled A
    __shared__ float Bs[128 * 17];   // 128 rows x 16 (+1 pad) of B

    const int lane = threadIdx.x;          // 0..31
    const int tm = blockIdx.x * 16;
    const int tn = blockIdx.y * 16;

    // ---- stage A tile (coalesced float4 global loads, fused relu+scale) ----
    for (int j = lane; j < 512; j += 32) {       // 512 float4 = 16*128 floats
        int row = j >> 5;                        // 32 float4 per row
        int kc  = (j & 31) << 2;
        int grow = tm + row;
        int gr = grow < N ? grow : N - 1;        // clamp (keeps EXEC full)
        const float4 v = *(const float4*)(X + (size_t)gr * K_DIM + kc);
        float s = scale[gr];
        float a0 = v.x, a1 = v.y, a2 = v.z, a3 = v.w;
        if (relu) {
            a0 = fmaxf(a0, 0.f); a1 = fmaxf(a1, 0.f);
            a2 = fmaxf(a2, 0.f); a3 = fmaxf(a3, 0.f);
        }
        float* p = As + row * 132 + kc;
        p[0] = a0 * s; p[1] = a1 * s; p[2] = a2 * s; p[3] = a3 * s;
    }
    // ---- stage B tile ----
    for (int j = lane; j < 512; j += 32) {       // 512 float4 = 128*16 floats
        int k  = j >> 2;                         // 4 float4 per row
        int nc = (j & 3) << 2;
        const float4 v = *(const float4*)(W + (size_t)k * Dout + tn + nc);
        float* p = Bs + k * 17 + nc;
        p[0] = v.x; p[1] = v.y; p[2] = v.z; p[3] = v.w;
    }
    __syncthreads();

    // ---- 32 x V_WMMA_F32_16X16X4_F32 over K=128 ----
    const int l   = lane & 15;
    const int sel = lane >> 4;                   // lanes 16-31 hold K+2,K+3
    v8f acc = {};
#pragma unroll
    for (int kk = 0; kk < K_DIM / 4; ++kk) {
        int kb = kk * 4 + sel * 2;
        v2f a; a.x = As[l * 132 + kb];  a.y = As[l * 132 + kb + 1];
        v2f b; b.x = Bs[kb * 17 + l];   b.y = Bs[(kb + 1) * 17 + l];
        acc = __builtin_amdgcn_wmma_f32_16x16x4_f32(
            /*neg_a=*/false, a, /*neg_b=*/false, b,
            /*c_mod=*/(short)0, acc, /*reuse_a=*/false, /*reuse_b=*/false);
    }

    // ---- store D: VGPR v -> row M = v + 8*sel, col N = l ----
#pragma unroll
    for (int v = 0; v < 8; ++v) {
        int row = tm + v + sel * 8;
        if (row < N) H[(size_t)row * Dout + tn + l] = acc[v];
    }
}

// ---------------- edge scatter: out[dst] += h[src] * rs_in[dst] ----------------
// blockDim = (D/4, 256/(D/4)); threadIdx.x = float4 chunk, threadIdx.y = edge-in-block
__global__ void scatter_edges(const float* __restrict__ h, const int* __restrict__ src,
                              const int* __restrict__ dst, const float* __restrict__ rs_in,
                              float* __restrict__ out, int E, int D) {
    int e = blockIdx.x * blockDim.y + threadIdx.y;
    if (e >= E) return;
    int c = threadIdx.x << 2;
    int s = src[e], d = dst[e];
    float w = rs_in[d];
    const float4 v = *(const float4*)(h + (size_t)s * D + c);
    float* o = out + (size_t)d * D + c;
    atomicAdd(o + 0, v.x * w);
    atomicAdd(o + 1, v.y * w);
    atomicAdd(o + 2, v.z * w);
    atomicAdd(o + 3, v.w * w);
}

extern "C" void kernel_launch(void* const* d_in, const int* in_sizes, int n_in,
                              void* d_out, int out_size, void* d_ws, size_t ws_size,
                              hipStream_t stream) {
    const float* x   = (const float*)d_in[0];
    const float* W1  = (const float*)d_in[1];
    const float* b1  = (const float*)d_in[2];
    const float* W2  = (const float*)d_in[3];
    const float* b2  = (const float*)d_in[4];
    const int*   src = (const int*)d_in[5];
    const int*   dst = (const int*)d_in[6];

    const int R  = 3;
    const int N  = in_sizes[0] / K_DIM;     // 100000
    const int E  = in_sizes[5] / R;         // 500000
    const int Dh = in_sizes[2] / R;         // 128
    const int Do = in_sizes[4] / R;         // 64

    float* ws = (float*)d_ws;
    size_t padRN = (((size_t)R * N) + 1023) & ~(size_t)1023;
    float* rs_out = ws;                         // [R,N] rsqrt(out-degree)
    float* rs_in  = ws + padRN;                 // [R,N] rsqrt(in-degree)
    float* h1     = ws + 2 * padRN;             // [N,128] layer-1 output
    float* htmp   = h1 + (size_t)N * Dh;        // [N,128] per-relation GEMM temp

    // --- degrees ---
    hipMemsetAsync(ws, 0, 2 * padRN * sizeof(float), stream);
    {
        dim3 g((E + 255) / 256, R);
        count_deg<<<g, 256, 0, stream>>>(src, dst, (int*)rs_out, (int*)rs_in, E, N);
        int n2 = (int)(2 * padRN);
        to_rsqrt<<<(n2 + 255) / 256, 256, 0, stream>>>((int*)ws, n2);
    }

    // --- layer 1: h1 = sum_r scatter( (x*rs_out_r) @ W1_r ) * rs_in_r + sum_r b1_r ---
    {
        long long tot = (long long)N * Dh;
        bias_init<<<(unsigned)((tot + 255) / 256), 256, 0, stream>>>(h1, b1, tot, Dh - 1, Dh);
        dim3 gg((N + 15) / 16, Dh / 16);
        dim3 sb(Dh / 4, 256 / (Dh / 4));
        dim3 sg((E + sb.y - 1) / sb.y);
        for (int r = 0; r < R; ++r) {
            gemm_wmma_f32<<<gg, 32, 0, stream>>>(
                x, W1 + (size_t)r * K_DIM * Dh, rs_out + (size_t)r * N, htmp, N, Dh, 0);
            scatter_edges<<<sg, sb, 0, stream>>>(
                htmp, src + (size_t)r * E, dst + (size_t)r * E,
                rs_in + (size_t)r * N, h1, E, Dh);
        }
    }

    // --- layer 2 (ReLU fused into GEMM A-load): d_out = sum_r ... + sum_r b2_r ---
    {
        float* out = (float*)d_out;
        long long tot = (long long)N * Do;
        bias_init<<<(unsigned)((tot + 255) / 256), 256, 0, stream>>>(out, b2, tot, Do - 1, Do);
        dim3 gg((N + 15) / 16, Do / 16);
        dim3 sb(Do / 4, 256 / (Do / 4));
        dim3 sg((E + sb.y - 1) / sb.y);
        for (int r = 0; r < R; ++r) {
            gemm_wmma_f32<<<gg, 32, 0, stream>>>(
                h1, W2 + (size_t)r * K_DIM * Do, rs_out + (size_t)r * N, htmp, N, Do, 1);
            scatter_edges<<<sg, sb, 0, stream>>>(
                htmp, src + (size_t)r * E, dst + (size_t)r * E,
                rs_in + (size_t)r * N, out, E, Do);
        }
    }
}